// SPDExpMap_33913061769781
// MI455X (gfx1250) — compile-verified
//
#include <hip/hip_runtime.h>

typedef __attribute__((ext_vector_type(2))) float v2f;
typedef __attribute__((ext_vector_type(8))) float v8f;

#define NDIM 64
#define MELEMS (NDIM * NDIM)   // 4096 floats = 16 KB

// 64x64x64 f32 matmul on LDS buffers using V_WMMA_F32_16X16X4_F32.
// D = A*B (+ diagAdd on the diagonal). 4 waves, 4 16x16 output tiles per wave.
__device__ __forceinline__ void matmul64(const float* __restrict__ A,
                                         const float* __restrict__ B,
                                         float* __restrict__ D,
                                         int lane, int wave, float diagAdd)
{
    const int col  = lane & 15;     // N (for B/C/D) and M (for A) within tile
    const int half = lane >> 4;     // 0: lanes 0-15, 1: lanes 16-31
    const int kOff = half * 2;      // A/B VGPR0 holds K=0 (lo lanes), K=2 (hi lanes)

#pragma unroll
    for (int t = 0; t < 4; ++t) {
        const int tile = wave * 4 + t;        // 16 tiles total
        const int mi = (tile >> 2) * 16;
        const int ni = (tile & 3) * 16;
        v8f acc = {};
#pragma unroll
        for (int kk = 0; kk < NDIM; kk += 4) {
            // A fragment: 16x4, lane holds row M=col, K = kk+kOff, kk+kOff+1
            const float* ap = A + (mi + col) * NDIM + kk + kOff;
            v2f a;
            a.x = ap[0];
            a.y = ap[1];
            // B fragment: 4x16, lane holds col N=col, K rows kk+kOff, kk+kOff+1
            const float* bp = B + (kk + kOff) * NDIM + ni + col;
            v2f b;
            b.x = bp[0];
            b.y = bp[NDIM];
            acc = __builtin_amdgcn_wmma_f32_16x16x4_f32(
                false, a, false, b, (short)0, acc, false, false);
        }
        // C/D layout: VGPR j holds M=j (lanes 0-15) and M=j+8 (lanes 16-31)
#pragma unroll
        for (int j = 0; j < 8; ++j) {
            const int row = mi + j + half * 8;
            const int c   = ni + col;
            float v = acc[j];
            if (row == c) v += diagAdd;
            D[row * NDIM + c] = v;
        }
    }
}

__global__ void __launch_bounds__(128)
spd_expmap_kernel(const float* __restrict__ in, float* __restrict__ out)
{
    __shared__ float Xs[MELEMS];
    __shared__ float Ps[MELEMS];
    __shared__ float Qs[MELEMS];
    __shared__ float red[4];

    const int tid  = threadIdx.x;
    const int lane = tid & 31;
    const int wave = tid >> 5;
    const size_t base = (size_t)blockIdx.x * MELEMS;

    // ---- load matrix (8 x float4 per thread, coalesced), Frobenius partial ----
    float ss = 0.f;
#pragma unroll
    for (int i = 0; i < 8; ++i) {
        const int idx = (i * 128 + tid) * 4;
        float4 v = *(const float4*)(in + base + idx);
        *(float4*)(Xs + idx) = v;
        ss += v.x * v.x + v.y * v.y + v.z * v.z + v.w * v.w;
    }
#pragma unroll
    for (int off = 16; off > 0; off >>= 1)
        ss += __shfl_xor(ss, off, 32);
    if (lane == 0) red[wave] = ss;
    __syncthreads();

    const float fnorm = sqrtf(red[0] + red[1] + red[2] + red[3]);

    // ---- scaling: ||X/2^k||_2 <= ||X||_F / 2^k <= theta ----
    const float theta = 0.35f;
    int k = 0;
    if (fnorm > theta) k = (int)ceilf(log2f(fnorm / theta));
    if (k < 0) k = 0;
    if (k > 24) k = 24;
    const float scale = exp2f((float)-k);

    // Taylor degree 8 coefficients 1/j!
    const float C8 = 2.4801587301587302e-05f;   // 1/40320
    const float C7 = 1.9841269841269841e-04f;   // 1/5040

    // scale X in place, init P = C8*X + C7*I (each thread touches its own elems)
#pragma unroll
    for (int i = 0; i < 8; ++i) {
        const int idx = (i * 128 + tid) * 4;
        const int row = idx >> 6;
        const int c0  = idx & 63;
        float4 v = *(const float4*)(Xs + idx);
        v.x *= scale; v.y *= scale; v.z *= scale; v.w *= scale;
        *(float4*)(Xs + idx) = v;
        float4 p;
        p.x = C8 * v.x + ((row == c0 + 0) ? C7 : 0.f);
        p.y = C8 * v.y + ((row == c0 + 1) ? C7 : 0.f);
        p.z = C8 * v.z + ((row == c0 + 2) ? C7 : 0.f);
        p.w = C8 * v.w + ((row == c0 + 3) ? C7 : 0.f);
        *(float4*)(Ps + idx) = p;
    }
    __syncthreads();

    // ---- Horner: P <- P*X + c_j*I, j = 6..0 (fully unrolled ping-pong) ----
    const float cs[7] = { 1.f / 720.f, 1.f / 120.f, 1.f / 24.f,
                          1.f / 6.f,   0.5f,        1.f, 1.f };
#pragma unroll
    for (int j = 0; j < 7; ++j) {
        const float* src = (j & 1) ? Qs : Ps;
        float* dst       = (j & 1) ? Ps : Qs;
        matmul64(src, Xs, dst, lane, wave, cs[j]);
        __syncthreads();
    }
    // result of Taylor poly now in Qs (7 writes: Qs,Ps,...,Qs)

    // ---- k squarings: R <- R*R (uniform trip count across the block) ----
    for (int i = 0; i < k; ++i) {
        if (i & 1) matmul64(Ps, Ps, Qs, lane, wave, 0.f);
        else       matmul64(Qs, Qs, Ps, lane, wave, 0.f);
        __syncthreads();
    }

    // final result: k even (incl. 0) -> Qs, k odd -> Ps
    if (k & 1) {
#pragma unroll
        for (int i = 0; i < 8; ++i) {
            const int idx = (i * 128 + tid) * 4;
            *(float4*)(out + base + idx) = *(const float4*)(Ps + idx);
        }
    } else {
#pragma unroll
        for (int i = 0; i < 8; ++i) {
            const int idx = (i * 128 + tid) * 4;
            *(float4*)(out + base + idx) = *(const float4*)(Qs + idx);
        }
    }
}

extern "C" void kernel_launch(void* const* d_in, const int* in_sizes, int n_in,
                              void* d_out, int out_size, void* d_ws, size_t ws_size,
                              hipStream_t stream) {
    const float* in = (const float*)d_in[0];
    float* out      = (float*)d_out;
    const int nmat  = in_sizes[0] / MELEMS;   // 16384
    spd_expmap_kernel<<<dim3(nmat), dim3(128), 0, stream>>>(in, out);
}